// CompactTensorSketch_79413945303746
// MI455X (gfx1250) — compile-verified
//
#include <hip/hip_runtime.h>
#include <stdint.h>

typedef float    v4f __attribute__((ext_vector_type(4)));
typedef uint32_t v4u __attribute__((ext_vector_type(4)));

#define BATCH   65536
#define DIM     768
#define EFF     1536
#define OUTD    2048
#define WAVES   4
#define THREADS (WAVES * 32)
#define NBLOCKS 1024

__global__ __launch_bounds__(THREADS)
void cts_sketch_kernel(const float* __restrict__ x,
                       const int*   __restrict__ hash_indices,
                       const int*   __restrict__ signs,
                       float*       __restrict__ out)
{
    // 12 KiB: packed (sign | bucket) control words, 4 per feature (b128 loads)
    __shared__ __align__(16) uint32_t ctrl[DIM * 4];
    // 72 KiB: per-wave private accumulators: 3 hashes x 1536 columns
    __shared__ __align__(16) float accum[WAVES][3 * EFF];
    // 24 KiB: per-wave double-buffered x-row staging (async global->LDS)
    __shared__ __align__(16) float xbuf[WAVES][2][DIM];

    const int tid  = threadIdx.x;
    const int wave = tid >> 5;
    const int lane = tid & 31;

    // ---- once per block: build control words ----------------------------
    for (int d = tid; d < DIM; d += THREADS) {
        #pragma unroll
        for (int i = 0; i < 3; ++i) {
            const uint32_t h = (uint32_t)hash_indices[i * DIM + d];
            const uint32_t s = (uint32_t)signs[i * DIM + d];
            // s==1 -> +1 (no flip), s==0 -> -1 (flip sign bit of the f32)
            ctrl[d * 4 + i] = (uint32_t)(i * EFF) + h + (s ? 0u : 0x80000000u);
        }
        ctrl[d * 4 + 3] = 0u;
    }

    const int gwave  = blockIdx.x * WAVES + wave;
    const int nwaves = gridDim.x * WAVES;
    float* acc = accum[wave];

    const uint32_t xoff0 = (uint32_t)(uintptr_t)&xbuf[wave][0][0];
    const uint32_t xoff1 = (uint32_t)(uintptr_t)&xbuf[wave][1][0];

    // async-copy one x row (3 KiB) into an LDS staging buffer: 6 x b128/lane
    auto prefetch = [&](int row, int buf) {
        uint64_t g = (uint64_t)(uintptr_t)(x + (size_t)row * DIM) + (uint32_t)(lane * 16);
        uint32_t l = (buf ? xoff1 : xoff0) + (uint32_t)(lane * 16);
        #pragma unroll
        for (int k = 0; k < 6; ++k) {
            asm volatile("global_load_async_to_lds_b128 %0, %1, off"
                         :: "v"(l), "v"(g) : "memory");
            g += 512; l += 512;
        }
    };

    int r = gwave;
    if (r < BATCH) prefetch(r, 0);   // warm the pipeline

    __syncthreads();                 // ctrl[] ready for all waves

    int buf = 0;
    for (; r < BATCH; r += nwaves) {
        // keep the async pipe full: fetch next row into the other buffer
        const int rn = r + nwaves;
        prefetch(rn < BATCH ? rn : r, buf ^ 1);

        // zero this wave's accumulators (overlaps with in-flight async copy)
        {
            const v4f z4 = {0.f, 0.f, 0.f, 0.f};
            #pragma unroll
            for (int j = 0; j < 36; ++j)
                *(v4f*)(acc + lane * 4 + j * 128) = z4;
        }

        // <=6 outstanding means the CURRENT row's 6 copies have landed
        asm volatile("s_wait_asynccnt 0x6" ::: "memory");

        // ---- 3-way signed scatter-add into LDS (ds_add_f32) -------------
        #pragma unroll
        for (int k = 0; k < 24; ++k) {
            const int d = lane + k * 32;
            const uint32_t xb = __float_as_uint(xbuf[wave][buf][d]);
            const v4u cw = *(const v4u*)(ctrl + d * 4);
            atomicAdd(&acc[cw.x & 0xffffu], __uint_as_float(xb ^ (cw.x & 0x80000000u)));
            atomicAdd(&acc[cw.y & 0xffffu], __uint_as_float(xb ^ (cw.y & 0x80000000u)));
            atomicAdd(&acc[cw.z & 0xffffu], __uint_as_float(xb ^ (cw.z & 0x80000000u)));
        }

        // ---- product of the 3 sketches + streaming (non-temporal) store -
        const size_t ob = (size_t)r * OUTD;
        #pragma unroll
        for (int j = 0; j < 12; ++j) {
            const int c = lane * 4 + j * 128;
            const v4f a0 = *(const v4f*)(acc + c);
            const v4f a1 = *(const v4f*)(acc + EFF + c);
            const v4f a2 = *(const v4f*)(acc + 2 * EFF + c);
            const v4f p  = a0 * a1 * a2;
            __builtin_nontemporal_store(p, (v4f*)(out + ob + c));
        }
        // zero-pad columns [1536, 2048)
        {
            const v4f z4 = {0.f, 0.f, 0.f, 0.f};
            #pragma unroll
            for (int j = 0; j < 4; ++j) {
                const int c = EFF + lane * 4 + j * 128;
                __builtin_nontemporal_store(z4, (v4f*)(out + ob + c));
            }
        }
        buf ^= 1;
    }
}

extern "C" void kernel_launch(void* const* d_in, const int* in_sizes, int n_in,
                              void* d_out, int out_size, void* d_ws, size_t ws_size,
                              hipStream_t stream) {
    const float* x  = (const float*)d_in[0];
    const int*   hi = (const int*)d_in[1];
    const int*   sg = (const int*)d_in[2];
    float*       o  = (float*)d_out;
    cts_sketch_kernel<<<NBLOCKS, THREADS, 0, stream>>>(x, hi, sg, o);
}